// Weighting_Module_72688026517959
// MI455X (gfx1250) — compile-verified
//
#include <hip/hip_runtime.h>
#include <hip/hip_bf16.h>
#include <math.h>

// ---------------------------------------------------------------------------
// Shapes / strides (compile-time constants for this problem instance)
// x: [B=8, C=32, T=16, H=112, W=112] f32
// ---------------------------------------------------------------------------
#define DB 8
#define DC 32
#define DT 16
#define DHW 12544            // 112*112
#define N_RED 401408         // C*H*W  (reduction size per saliency map)
#define XB_STR 6422528       // C*T*HW
#define XC_STR 200704        // T*HW
#define XT_STR 12544         // HW
#define GB_STR 401408        // C*HW
#define F4_PER_ROW 3136      // HW/4
#define ALPHA 0.3f
#define FLT_BIG 3.402823466e38f

// workspace layout (float offsets)
#define WS_G    ((size_t)0)            // 3,211,264 floats : g = mean over T
#define WS_P1   ((size_t)3211264)      // 4096*4 : per (b,t,c) sum,sumsq,min,max
#define WS_SBT  ((size_t)3227648)      // 128*4  : per (b,t)  sum,sumsq,min,max
#define WS_GST  ((size_t)3228160)      // 8*4    : per b stats of g
#define WS_P2   ((size_t)3228192)      // 4096*2 : per (b,t,c) cross,sim
#define WS_W    ((size_t)3236384)      // 128    : final weights w[b,t]

typedef __attribute__((ext_vector_type(2))) float v2f;
typedef __attribute__((ext_vector_type(8))) float v8f;
typedef __attribute__((ext_vector_type(4))) int   v4i;

#if __has_builtin(__builtin_amdgcn_wmma_f32_16x16x4_f32)
#define HAVE_WMMA_F32X4 1
#endif
#if __has_builtin(__builtin_amdgcn_global_load_async_to_lds_b128)
#define HAVE_ASYNC_LDS 1
#endif

// ---------------------------------------------------------------------------
// K1: g[b,c,hw] = (1/16) * sum_t x[b,c,t,hw]
// One wave per 16-wide hw tile. WMMA path: D = A x ones, chained over 4
// k-chunks, so D[m,*] = sum over all 16 t for hw position m. With B == all
// ones the result is independent of B's register layout; A and D layouts are
// the ISA-documented 16x4 / 16x16 f32 layouts.
// grid: 25088 x 256  (8 waves/block, 200704 waves == 200704 tiles)
// ---------------------------------------------------------------------------
__global__ void k1_meanT(const float* __restrict__ x, float* __restrict__ g) {
  const int wave = blockIdx.x * (blockDim.x >> 5) + (threadIdx.x >> 5);
  const int lane = threadIdx.x & 31;
  const int j  = wave % 784;          // hw tile within (b,c)
  const int bc = wave / 784;          // b*32 + c
  const float* base = x + (size_t)bc * XC_STR + (size_t)j * 16;
  float* gout = g + (size_t)bc * DHW + (size_t)j * 16;
#if HAVE_WMMA_F32X4
  const int m = lane & 15;            // row (hw offset)
  const int kh = lane >> 4;           // lanes 16-31 hold K=2,3 of each chunk
  v8f acc = {};
  v2f bones; bones.x = 1.0f; bones.y = 1.0f;
#pragma unroll
  for (int kc = 0; kc < 4; ++kc) {
    const int t0 = kc * 4 + kh * 2;
    v2f a;
    a.x = base[(size_t)t0 * XT_STR + m];
    a.y = base[(size_t)(t0 + 1) * XT_STR + m];
    acc = __builtin_amdgcn_wmma_f32_16x16x4_f32(false, a, false, bones,
                                                (short)0, acc, false, false);
  }
  // D layout: VGPR v, lanes 0-15 -> row v ; lanes 16-31 -> row v+8 (all cols equal)
  if ((lane & 15) < 8) {
    float val = acc[0];
#pragma unroll
    for (int v = 1; v < 8; ++v)
      if ((lane & 7) == v) val = acc[v];
    const int row = (lane & 7) + ((lane >> 4) << 3);
    gout[row] = val * 0.0625f;
  }
#else
  if (lane < 16) {
    float s = 0.0f;
#pragma unroll
    for (int t = 0; t < DT; ++t) s += base[(size_t)t * XT_STR + lane];
    gout[lane] = s * 0.0625f;
  }
#endif
}

// ---------------------------------------------------------------------------
// K2: per (b,t,c) stats of x slice (12544 contiguous floats):
//     sum, sumsq, min, max -> part1[p*4..]
// grid: 4096 x 256
// ---------------------------------------------------------------------------
__global__ void k2_stats_x(const float* __restrict__ x, float* __restrict__ part1) {
  __shared__ float s1[256], s2[256], s3[256], s4[256];
  const int p = blockIdx.x;                  // ((b*T+t)*C + c)
  const int b = p >> 9;
  const int t = (p >> 5) & 15;
  const int c = p & 31;
  const float4* xv = (const float4*)(x + (size_t)b * XB_STR +
                                         (size_t)c * XC_STR + (size_t)t * XT_STR);
  float sum = 0.f, sq = 0.f, mn = FLT_BIG, mx = -FLT_BIG;
  for (int i = threadIdx.x; i < F4_PER_ROW; i += 256) {
    __builtin_prefetch(xv + i + 256, 0, 1);
    float4 v = xv[i];
    sum += (v.x + v.y) + (v.z + v.w);
    sq  += v.x * v.x + v.y * v.y + v.z * v.z + v.w * v.w;
    mn = fminf(mn, fminf(fminf(v.x, v.y), fminf(v.z, v.w)));
    mx = fmaxf(mx, fmaxf(fmaxf(v.x, v.y), fmaxf(v.z, v.w)));
  }
  const int tid = threadIdx.x;
  s1[tid] = sum; s2[tid] = sq; s3[tid] = mn; s4[tid] = mx;
  __syncthreads();
  for (int off = 128; off > 0; off >>= 1) {
    if (tid < off) {
      s1[tid] += s1[tid + off];
      s2[tid] += s2[tid + off];
      s3[tid] = fminf(s3[tid], s3[tid + off]);
      s4[tid] = fmaxf(s4[tid], s4[tid + off]);
    }
    __syncthreads();
  }
  if (tid == 0) {
    part1[(size_t)p * 4 + 0] = s1[0];
    part1[(size_t)p * 4 + 1] = s2[0];
    part1[(size_t)p * 4 + 2] = s3[0];
    part1[(size_t)p * 4 + 3] = s4[0];
  }
}

// ---------------------------------------------------------------------------
// K3: per-b stats of g (401408 floats per b). grid: 8 x 256
// ---------------------------------------------------------------------------
__global__ void k3_stats_g(const float* __restrict__ g, float* __restrict__ gst) {
  __shared__ float s1[256], s2[256], s3[256], s4[256];
  const int b = blockIdx.x;
  const float4* gv = (const float4*)(g + (size_t)b * GB_STR);
  float sum = 0.f, sq = 0.f, mn = FLT_BIG, mx = -FLT_BIG;
  for (int i = threadIdx.x; i < (GB_STR / 4); i += 256) {
    __builtin_prefetch(gv + i + 256, 0, 1);
    float4 v = gv[i];
    sum += (v.x + v.y) + (v.z + v.w);
    sq  += v.x * v.x + v.y * v.y + v.z * v.z + v.w * v.w;
    mn = fminf(mn, fminf(fminf(v.x, v.y), fminf(v.z, v.w)));
    mx = fmaxf(mx, fmaxf(fmaxf(v.x, v.y), fmaxf(v.z, v.w)));
  }
  const int tid = threadIdx.x;
  s1[tid] = sum; s2[tid] = sq; s3[tid] = mn; s4[tid] = mx;
  __syncthreads();
  for (int off = 128; off > 0; off >>= 1) {
    if (tid < off) {
      s1[tid] += s1[tid + off];
      s2[tid] += s2[tid + off];
      s3[tid] = fminf(s3[tid], s3[tid + off]);
      s4[tid] = fmaxf(s4[tid], s4[tid + off]);
    }
    __syncthreads();
  }
  if (tid == 0) {
    gst[b * 4 + 0] = s1[0];
    gst[b * 4 + 1] = s2[0];
    gst[b * 4 + 2] = s3[0];
    gst[b * 4 + 3] = s4[0];
  }
}

// ---------------------------------------------------------------------------
// K4: reduce part1 over c -> per (b,t) stats. grid: 1 x 128
// ---------------------------------------------------------------------------
__global__ void k4_combine_stats(const float* __restrict__ part1, float* __restrict__ sbt) {
  const int j = threadIdx.x;                 // b*16 + t
  float sum = 0.f, sq = 0.f, mn = FLT_BIG, mx = -FLT_BIG;
  for (int c = 0; c < DC; ++c) {
    const float* q = part1 + ((size_t)j * DC + c) * 4;
    sum += q[0]; sq += q[1];
    mn = fminf(mn, q[2]); mx = fmaxf(mx, q[3]);
  }
  sbt[j * 4 + 0] = sum;
  sbt[j * 4 + 1] = sq;
  sbt[j * 4 + 2] = mn;
  sbt[j * 4 + 3] = mx;
}

// ---------------------------------------------------------------------------
// K5: per (b,t,c): cross = sum(x*g), sim = sum(min(sn,gn)).
// g row (49 KB) staged into LDS via CDNA5 async-to-LDS, then x streamed.
// grid: 4096 x 256
// ---------------------------------------------------------------------------
__global__ void k5_cross_sim(const float* __restrict__ x, const float* __restrict__ g,
                             const float* __restrict__ sbt, const float* __restrict__ gst,
                             float* __restrict__ part2) {
  __shared__ float gl[DHW];
  __shared__ float r1[256], r2[256];
  const int p = blockIdx.x;
  const int b = p >> 9;
  const int t = (p >> 5) & 15;
  const int c = p & 31;
  const float* gsrc = g + ((size_t)b * DC + c) * DHW;

#if HAVE_ASYNC_LDS
  for (int i = threadIdx.x; i < F4_PER_ROW; i += 256) {
    __builtin_amdgcn_global_load_async_to_lds_b128(
        (__attribute__((address_space(1))) v4i*)(gsrc + (size_t)i * 4),
        (__attribute__((address_space(3))) v4i*)(gl + (size_t)i * 4), 0, 0);
  }
#if __has_builtin(__builtin_amdgcn_s_wait_asynccnt)
  __builtin_amdgcn_s_wait_asynccnt(0);
#else
  asm volatile("s_wait_asynccnt 0" ::: "memory");
#endif
  __syncthreads();
#else
  for (int i = threadIdx.x; i < F4_PER_ROW; i += 256)
    ((float4*)gl)[i] = ((const float4*)gsrc)[i];
  __syncthreads();
#endif

  const int bt = b * DT + t;
  const float ssum = sbt[bt * 4 + 0];
  const float smn  = sbt[bt * 4 + 2];
  const float gsum = gst[b * 4 + 0];
  const float gmn  = gst[b * 4 + 2];
  const float invDs = 1.0f / (ssum - (float)N_RED * smn);
  const float invDg = 1.0f / (gsum - (float)N_RED * gmn);

  const float4* xv = (const float4*)(x + (size_t)b * XB_STR +
                                         (size_t)c * XC_STR + (size_t)t * XT_STR);
  float cross = 0.f, sim = 0.f;
  for (int i = threadIdx.x; i < F4_PER_ROW; i += 256) {
    __builtin_prefetch(xv + i + 256, 0, 1);
    float4 xw = xv[i];
    float4 gw = ((const float4*)gl)[i];
    cross += xw.x * gw.x + xw.y * gw.y + xw.z * gw.z + xw.w * gw.w;
    sim += fminf((xw.x - smn) * invDs, (gw.x - gmn) * invDg);
    sim += fminf((xw.y - smn) * invDs, (gw.y - gmn) * invDg);
    sim += fminf((xw.z - smn) * invDs, (gw.z - gmn) * invDg);
    sim += fminf((xw.w - smn) * invDs, (gw.w - gmn) * invDg);
  }
  const int tid = threadIdx.x;
  r1[tid] = cross; r2[tid] = sim;
  __syncthreads();
  for (int off = 128; off > 0; off >>= 1) {
    if (tid < off) { r1[tid] += r1[tid + off]; r2[tid] += r2[tid + off]; }
    __syncthreads();
  }
  if (tid == 0) {
    part2[(size_t)p * 2 + 0] = r1[0];
    part2[(size_t)p * 2 + 1] = r2[0];
  }
}

// ---------------------------------------------------------------------------
// K6: finalize 128 scores -> weights. grid: 1 x 128
// ---------------------------------------------------------------------------
__global__ void k6_weights(const float* __restrict__ part2, const float* __restrict__ sbt,
                           const float* __restrict__ gst, float* __restrict__ wbt) {
  __shared__ float sh[128];
  const int j = threadIdx.x;                 // b*16 + t
  const int b = j >> 4;
  float cross = 0.f, sim = 0.f;
  for (int c = 0; c < DC; ++c) {
    cross += part2[((size_t)j * DC + c) * 2 + 0];
    sim   += part2[((size_t)j * DC + c) * 2 + 1];
  }
  const float n = (float)N_RED, nm1 = (float)(N_RED - 1);
  const float ssum = sbt[j * 4 + 0], ssq = sbt[j * 4 + 1];
  const float gsum = gst[b * 4 + 0], gsq = gst[b * 4 + 1];
  const float smean = ssum / n, gmean = gsum / n;
  const float sstd = sqrtf((ssq - ssum * ssum / n) / nm1);
  const float gstd = sqrtf((gsq - gsum * gsum / n) / nm1);
  const float cc = (cross - n * smean * gmean) / (nm1 * sstd * gstd);
  const float score = ALPHA * cc + (1.0f - ALPHA) * sim;
  const float s = 1.0f - score;
  sh[j] = s;
  __syncthreads();
  float mx = sh[b * 16];
  for (int tt = 1; tt < DT; ++tt) mx = fmaxf(mx, sh[b * 16 + tt]);
  wbt[j] = expf(s / mx);
}

// ---------------------------------------------------------------------------
// K7: out = x * w[b,t].  grid: 50176 x 256 (exactly 12,845,056 float4)
// ---------------------------------------------------------------------------
__global__ void k7_scale(const float* __restrict__ x, const float* __restrict__ wbt,
                         float* __restrict__ out) {
  const size_t q = (size_t)blockIdx.x * 256 + threadIdx.x;   // float4 index
  const int r = (int)(q / F4_PER_ROW);       // row = (b*32+c)*16 + t
  const int t = r & 15;
  const int b = r >> 9;
  const float w = wbt[b * DT + t];
  float4 v = ((const float4*)x)[q];
  v.x *= w; v.y *= w; v.z *= w; v.w *= w;
  ((float4*)out)[q] = v;
}

// ---------------------------------------------------------------------------
extern "C" void kernel_launch(void* const* d_in, const int* in_sizes, int n_in,
                              void* d_out, int out_size, void* d_ws, size_t ws_size,
                              hipStream_t stream) {
  const float* x = (const float*)d_in[0];
  float* out = (float*)d_out;
  float* ws = (float*)d_ws;
  float* g    = ws + WS_G;
  float* p1   = ws + WS_P1;
  float* sbt  = ws + WS_SBT;
  float* gst  = ws + WS_GST;
  float* p2   = ws + WS_P2;
  float* wbt  = ws + WS_W;

  k1_meanT     <<<25088, 256, 0, stream>>>(x, g);
  k2_stats_x   <<<4096,  256, 0, stream>>>(x, p1);
  k3_stats_g   <<<8,     256, 0, stream>>>(g, gst);
  k4_combine_stats<<<1,  128, 0, stream>>>(p1, sbt);
  k5_cross_sim <<<4096,  256, 0, stream>>>(x, g, sbt, gst, p2);
  k6_weights   <<<1,     128, 0, stream>>>(p2, sbt, gst, wbt);
  k7_scale     <<<50176, 256, 0, stream>>>(x, wbt, out);
}